// SplitTableBatchedEmbeddingBagsCodegen_56556129354008
// MI455X (gfx1250) — compile-verified
//
#include <hip/hip_runtime.h>
#include <stdint.h>

typedef unsigned int v4u __attribute__((ext_vector_type(4)));
typedef unsigned int v8u __attribute__((ext_vector_type(8)));
typedef int          v4i __attribute__((ext_vector_type(4)));
typedef float        v4f __attribute__((ext_vector_type(4)));

namespace {
constexpr int kT = 8;        // tables
constexpr int kE = 250000;   // rows per table
constexpr int kD = 128;      // embedding dim
constexpr int kB = 4096;     // bags
constexpr int kL = 32;       // indices per bag
constexpr int kRowsPerTdm = 8;                 // 32-bit gather indices -> 8 rows per D#
constexpr int kChunks     = kL / kRowsPerTdm;  // 4 TDM ops per bag
constexpr int kRowBytes   = kD * 4;            // 512 B
constexpr int kChunkBytes = kRowsPerTdm * kRowBytes; // 4 KB
}

// One wave32 workgroup per (table, bag). TDM gather-mode pulls 8 embedding rows
// per tensor_load_to_lds; 4 in-flight descriptors are drained in order with
// s_wait_tensorcnt 3/2/1/0 while accumulating from LDS.
__global__ __launch_bounds__(32)
void emb_bag_tdm_gather(const float* __restrict__ weights,
                        const int* __restrict__ indices,
                        float* __restrict__ out)
{
    __shared__ __align__(16) float tile[kChunks * kRowsPerTdm * kD]; // 16 KB

    const int bag  = (int)blockIdx.x;     // 0 .. T*B-1
    const int t    = bag >> 12;           // bag / 4096
    const int b    = bag & (kB - 1);
    const int lane = (int)threadIdx.x;    // 0..31

    const int* bip = indices + ((size_t)t * kB + b) * kL;
    const unsigned long long tbl =
        (unsigned long long)(uintptr_t)(weights + (size_t)t * kE * kD);
    // Generic LDS pointer: low 32 bits are the wave-relative LDS byte address.
    const unsigned ldsBase = (unsigned)(uintptr_t)(void*)&tile[0];

    // ---- D# group 1 (256b): constant across chunks -------------------------
    // [17:16] data_size=2 (4B); [79:48] tensor_dim0=128; [111:80] tensor_dim1=E;
    // [127:112] tile_dim0=128; [143:128] tile_dim1=8 (#valid gather indices);
    // [207:160] tensor_dim0_stride=128. Everything else zero (no pad/iterate/
    // multicast/atomic-barrier; dim1_stride ignored in gather mode).
    v8u g1;
    g1[0] = 2u << 16;
    g1[1] = ((unsigned)kD & 0xFFFFu) << 16;
    g1[2] = (((unsigned)kD >> 16) & 0xFFFFu) | (((unsigned)kE & 0xFFFFu) << 16);
    g1[3] = (((unsigned)kE >> 16) & 0xFFFFu) | ((unsigned)kD << 16);
    g1[4] = (unsigned)kRowsPerTdm;
    g1[5] = (unsigned)kD;
    g1[6] = 0u;
    g1[7] = 0u;

    // ---- issue all 4 gather descriptors (TENSORcnt -> 4) -------------------
#pragma unroll
    for (int c = 0; c < kChunks; ++c) {
        v4u g0;
        g0[0] = 0xC0000001u;  // count=1 | gather_index_size=32b | gather_mode=1
        g0[1] = ldsBase + (unsigned)(c * kChunkBytes);           // lds_addr
        g0[2] = (unsigned)tbl;                                   // global_addr lo
        g0[3] = (unsigned)((tbl >> 32) & 0x01FFFFFFu) | (2u << 30); // hi | type=2

        v4i g2, g3;  // row indices 0..7 (32-bit index mode)
        g2[0] = __builtin_amdgcn_readfirstlane(bip[c * 8 + 0]);
        g2[1] = __builtin_amdgcn_readfirstlane(bip[c * 8 + 1]);
        g2[2] = __builtin_amdgcn_readfirstlane(bip[c * 8 + 2]);
        g2[3] = __builtin_amdgcn_readfirstlane(bip[c * 8 + 3]);
        g3[0] = __builtin_amdgcn_readfirstlane(bip[c * 8 + 4]);
        g3[1] = __builtin_amdgcn_readfirstlane(bip[c * 8 + 5]);
        g3[2] = __builtin_amdgcn_readfirstlane(bip[c * 8 + 6]);
        g3[3] = __builtin_amdgcn_readfirstlane(bip[c * 8 + 7]);

        asm volatile("tensor_load_to_lds %0, %1, %2, %3"
                     :
                     : "s"(g0), "s"(g1), "s"(g2), "s"(g3)
                     : "memory");
    }

    // ---- pipelined drain: TDM ops complete in order ------------------------
    v4f acc;
    acc[0] = acc[1] = acc[2] = acc[3] = 0.0f;
#pragma unroll
    for (int c = 0; c < kChunks; ++c) {
        asm volatile("s_wait_tensorcnt %0" :: "i"(kChunks - 1 - c) : "memory");
#pragma unroll
        for (int r = 0; r < kRowsPerTdm; ++r) {
            const v4f* p =
                (const v4f*)&tile[(c * kRowsPerTdm + r) * kD + lane * 4];
            acc += *p;  // ds_load_b128 + packed adds
        }
    }

    // out[b, t*D + lane*4 .. +3], coalesced global_store_b128
    v4f* op = (v4f*)(out + (size_t)b * (kT * kD) + (size_t)t * kD + (size_t)lane * 4);
    *op = acc;
}

extern "C" void kernel_launch(void* const* d_in, const int* in_sizes, int n_in,
                              void* d_out, int out_size, void* d_ws, size_t ws_size,
                              hipStream_t stream) {
    (void)in_sizes; (void)n_in; (void)out_size; (void)d_ws; (void)ws_size;
    const float* weights = (const float*)d_in[0];   // [T, E, D] float32
    const int*   indices = (const int*)d_in[1];     // [T, B, L] int
    float*       out     = (float*)d_out;           // [B, T*D] float32

    dim3 grid(kT * kB);   // one wave32 workgroup per bag
    dim3 block(32);
    emb_bag_tdm_gather<<<grid, block, 0, stream>>>(weights, indices, out);
}